// DPDC_and_CE_loss_15169824489502
// MI455X (gfx1250) — compile-verified
//
#include <hip/hip_runtime.h>

// ---------------------------------------------------------------------------
// DPDC + CE loss, gfx1250 (MI455X).
// Bandwidth-bound streaming reduction (~118 MB @ 23.3 TB/s => ~5us floor).
// Block-level reduction done with V_WMMA_F32_16X16X4_F32 (ones-matrix GEMM:
// D[m][n] += sum_k B[k][n]), keeping full f32 precision on the CDNA5 matrix
// pipe instead of 5-step shuffle trees.
// ---------------------------------------------------------------------------

typedef float v2f __attribute__((ext_vector_type(2)));
typedef float v8f __attribute__((ext_vector_type(8)));

#define NCLS   5
#define SVOL   2097152   // 128^3 spatial elements per batch
#define QTOT   524288    // SVOL / 4 (float4 quads)
#define BX     512       // grid.x
#define TPB    256       // threads per block (8 waves, wave32)
#define NACC   16        // 5 sumP + 5 tp + 5 cnt + 1 nll
#define LDSPAD 17        // row pad -> bank-conflict-free (17 odd)

__global__ void zero_ws(float* ws) {
    if (threadIdx.x < 32) ws[threadIdx.x] = 0.0f;
}

__global__ __launch_bounds__(TPB) void loss_main(
    const float* __restrict__ net, const int* __restrict__ tgt,
    const float* __restrict__ dist, float* __restrict__ ws) {
    const int tid = threadIdx.x;
    const int b   = blockIdx.y;

    const float* __restrict__ netb  = net  + (size_t)b * NCLS * SVOL;
    const int*   __restrict__ tgtb  = tgt  + (size_t)b * SVOL;
    const float* __restrict__ distb = dist + (size_t)b * SVOL;

    float acc[NACC];
#pragma unroll
    for (int i = 0; i < NACC; ++i) acc[i] = 0.0f;

    // Uniform trip count (QTOT / (BX*TPB) == 4) -> no divergence, EXEC all-1s.
    for (int q = blockIdx.x * TPB + tid; q < QTOT; q += BX * TPB) {
        const int s = q * 4;
        const float4 x0 = *(const float4*)(netb + 0 * SVOL + s);
        const float4 x1 = *(const float4*)(netb + 1 * SVOL + s);
        const float4 x2 = *(const float4*)(netb + 2 * SVOL + s);
        const float4 x3 = *(const float4*)(netb + 3 * SVOL + s);
        const float4 x4 = *(const float4*)(netb + 4 * SVOL + s);
        const int4   gv = *(const int4*)(tgtb + s);
        const float4 dv = *(const float4*)(distb + s);

#define PROC(CMP, GT, DD) do {                                               \
        const float l0 = x0.CMP, l1 = x1.CMP, l2 = x2.CMP,                   \
                    l3 = x3.CMP, l4 = x4.CMP;                                \
        const int   gt = (GT);                                               \
        const float dd = (DD);                                               \
        const float m  = fmaxf(fmaxf(fmaxf(l0, l1), fmaxf(l2, l3)), l4);     \
        const float e0 = __expf(l0 - m), e1 = __expf(l1 - m);                \
        const float e2 = __expf(l2 - m), e3 = __expf(l3 - m);                \
        const float e4 = __expf(l4 - m);                                     \
        const float zs = e0 + e1 + e2 + e3 + e4;                             \
        const float iv = 1.0f / zs;                                          \
        acc[0] += e0 * iv; acc[1] += e1 * iv; acc[2] += e2 * iv;             \
        acc[3] += e3 * iv; acc[4] += e4 * iv;                                \
        const float xg = (gt == 0) ? l0 : (gt == 1) ? l1 :                   \
                         (gt == 2) ? l2 : (gt == 3) ? l3 : l4;               \
        const float eg = (gt == 0) ? e0 : (gt == 1) ? e1 :                   \
                         (gt == 2) ? e2 : (gt == 3) ? e3 : e4;               \
        acc[15] += m + __logf(zs) - xg;        /* NLL */                     \
        const float pd = eg * iv * dd;                                       \
        acc[5]  += (gt == 0) ? pd : 0.0f;  acc[10] += (gt == 0) ? 1.f : 0.f; \
        acc[6]  += (gt == 1) ? pd : 0.0f;  acc[11] += (gt == 1) ? 1.f : 0.f; \
        acc[7]  += (gt == 2) ? pd : 0.0f;  acc[12] += (gt == 2) ? 1.f : 0.f; \
        acc[8]  += (gt == 3) ? pd : 0.0f;  acc[13] += (gt == 3) ? 1.f : 0.f; \
        acc[9]  += (gt == 4) ? pd : 0.0f;  acc[14] += (gt == 4) ? 1.f : 0.f; \
    } while (0)

        PROC(x, gv.x, dv.x);
        PROC(y, gv.y, dv.y);
        PROC(z, gv.z, dv.z);
        PROC(w, gv.w, dv.w);
#undef PROC
    }

    // ---- Block reduction: 256x16 column-sum via ones-matrix WMMA ----------
    __shared__ float lds[TPB * LDSPAD];
#pragma unroll
    for (int j = 0; j < NACC; ++j) lds[tid * LDSPAD + j] = acc[j];
    __syncthreads();

    if (tid < 32) {   // wave 0 only: full wave => EXEC all-1s for WMMA
        const int n = tid & 15;          // column (N)
        const int h = tid >> 4;          // lane-half selects K=2h..2h+1
        v2f A; A.x = 1.0f; A.y = 1.0f;   // ones(16x4)
        v8f D = {};                      // C accumulator starts at 0
        for (int row = 0; row < TPB; row += 4) {
            v2f B;
            B.x = lds[(row + 2 * h + 0) * LDSPAD + n];  // B[2h+0][n]
            B.y = lds[(row + 2 * h + 1) * LDSPAD + n];  // B[2h+1][n]
            // D[m][n] += sum_k B[k][n]  (every row m equals colSum)
            D = __builtin_amdgcn_wmma_f32_16x16x4_f32(
                    false, A, false, B, (short)0, D, false, false);
        }
        // D row M=0 lives in VGPR0 of lanes 0..15 (N = lane).
        if (tid < 16) atomicAdd(&ws[b * NACC + n], D[0]);
    }
}

__global__ void finalize(const float* __restrict__ ws, float* __restrict__ out) {
    if (threadIdx.x != 0) return;
    const float SM = 1e-5f;
    float dcsum = 0.0f, nll = 0.0f;
    for (int b = 0; b < 2; ++b) {
        for (int c = 0; c < NCLS; ++c) {
            const float sp  = ws[b * NACC + c];          // sum p
            const float tp  = ws[b * NACC + 5 + c];      // sum p*onehot*dist
            const float cnt = ws[b * NACC + 10 + c];     // sum onehot
            dcsum += (2.0f * tp + SM) / (sp + cnt + SM);
        }
        nll += ws[b * NACC + 15];
    }
    const float dpdc = -dcsum / 10.0f;                   // mean over (B,C)
    const float ce   = nll / 4194304.0f;                 // mean over 2*128^3
    out[0] = ce + dpdc;
}

extern "C" void kernel_launch(void* const* d_in, const int* in_sizes, int n_in,
                              void* d_out, int out_size, void* d_ws, size_t ws_size,
                              hipStream_t stream) {
    const float* net  = (const float*)d_in[0];
    const int*   tgt  = (const int*)d_in[1];
    const float* dist = (const float*)d_in[2];
    float*       out  = (float*)d_out;
    float*       ws   = (float*)d_ws;

    zero_ws<<<1, 32, 0, stream>>>(ws);
    loss_main<<<dim3(BX, 2), TPB, 0, stream>>>(net, tgt, dist, ws);
    finalize<<<1, 64, 0, stream>>>(ws, out);
}